// MyModule_20031727468693
// MI455X (gfx1250) — compile-verified
//
#include <hip/hip_runtime.h>

// ---------------------------------------------------------------------------
// Sparse 3x3 conv (non-submanifold), CIN=2, COUT=5, out = dense 4x1022x1022x5.
// Strategy:
//   Kernel 1: zero-fill the 83.6 MB output (float4 vectorized, memory bound).
//   Kernel 2: per wave, 16 points -> 3x V_WMMA_F32_16X16X4_F32 computing all
//             9 taps x 5 channels (15 of 16 N-columns used per WMMA, K=CIN
//             zero-padded 2->4), then scatter with global_atomic_add_f32.
// ---------------------------------------------------------------------------

typedef __attribute__((ext_vector_type(2))) float v2f;
typedef __attribute__((ext_vector_type(8))) float v8f;

#define SC_B    4
#define SC_HOUT 1022
#define SC_WOUT 1022
#define SC_COUT 5

__global__ void sc_zero_kernel(float* __restrict__ out, long n) {
    const long n4 = n >> 2;
    float4* __restrict__ o4 = (float4*)out;
    const long stride = (long)gridDim.x * blockDim.x;
    const long tid = (long)blockIdx.x * blockDim.x + threadIdx.x;
    const float4 z = make_float4(0.f, 0.f, 0.f, 0.f);
    for (long i = tid; i < n4; i += stride) o4[i] = z;
    // scalar tail (out_size here is divisible by 4, but be safe)
    const long rem = n & 3;
    if (tid < rem) out[(n4 << 2) + tid] = 0.f;
}

__launch_bounds__(256)
__global__ void sc_conv_wmma_kernel(const float* __restrict__ features,
                                    const int*   __restrict__ indices,
                                    const float* __restrict__ weight,
                                    float*       __restrict__ out,
                                    int N) {
    const int lane  = threadIdx.x & 31;
    const int wave  = threadIdx.x >> 5;
    const int pbase = (blockIdx.x * 8 + wave) * 16;

    const int ncol = lane & 15;   // N column of the 16x16 D tile
    const int hi   = lane >> 4;   // 0: lanes 0-15, 1: lanes 16-31

    // ---- B matrices (weights), one per ky in {0,1,2} ----
    // Column n = kx*5 + c  (n < 15), row k = input channel (k>=2 is zero pad).
    // weight flat layout (K,K,CIN,COUT): w[ky][kx][ci][c] at (ky*3+kx)*10 + ci*5 + c
    int c = 0, kx = 0;
    if (ncol < 15) { kx = ncol / 5; c = ncol - kx * 5; }
    v2f b0 = {0.f, 0.f}, b1 = {0.f, 0.f}, b2 = {0.f, 0.f};
    if (hi == 0 && ncol < 15) {
        const int o0 = (0 * 3 + kx) * 10 + c;
        const int o1 = (1 * 3 + kx) * 10 + c;
        const int o2 = (2 * 3 + kx) * 10 + c;
        b0.x = weight[o0]; b0.y = weight[o0 + 5];
        b1.x = weight[o1]; b1.y = weight[o1 + 5];
        b2.x = weight[o2]; b2.y = weight[o2 + 5];
    }

    // ---- A matrix: 16 points' features (16 x CIN, K padded to 4) ----
    v2f a = {0.f, 0.f};
    if (hi == 0) {
        const int p = pbase + lane;
        if (p < N) {
            a.x = features[2 * p];
            a.y = features[2 * p + 1];
        }
    }

    // ---- 3 WMMAs: D_j[m][kx*5+c] = contribution of point m for tap (ky=j,kx), ch c
    // Uniform control flow here: EXEC is all ones as required by WMMA.
    v8f zc = {};
    v8f d0 = __builtin_amdgcn_wmma_f32_16x16x4_f32(false, a, false, b0, (short)0, zc, false, false);
    v8f d1 = __builtin_amdgcn_wmma_f32_16x16x4_f32(false, a, false, b1, (short)0, zc, false, false);
    v8f d2 = __builtin_amdgcn_wmma_f32_16x16x4_f32(false, a, false, b2, (short)0, zc, false, false);

    if (ncol >= 15) return;  // column 15 is padding

    // ---- Scatter: D VGPR v holds row M = v + hi*8, column ncol ----
    #pragma unroll
    for (int v = 0; v < 8; ++v) {
        const int m = v + hi * 8;
        const int p = pbase + m;
        if (p >= N) continue;
        const int bi = indices[3 * p + 0];
        const int y  = indices[3 * p + 1];
        const int x  = indices[3 * p + 2];
        const int xo = x - kx;
        if (xo < 0 || xo >= SC_WOUT) continue;
        const float vals[3] = { d0[v], d1[v], d2[v] };
        #pragma unroll
        for (int ky = 0; ky < 3; ++ky) {
            const int yo = y - ky;
            if (yo >= 0 && yo < SC_HOUT) {
                float* addr = out +
                    ((((long)bi * SC_HOUT + yo) * SC_WOUT + xo) * SC_COUT + c);
                __hip_atomic_fetch_add(addr, vals[ky], __ATOMIC_RELAXED,
                                       __HIP_MEMORY_SCOPE_AGENT);
            }
        }
    }
}

extern "C" void kernel_launch(void* const* d_in, const int* in_sizes, int n_in,
                              void* d_out, int out_size, void* d_ws, size_t ws_size,
                              hipStream_t stream) {
    const float* features = (const float*)d_in[0];
    const int*   indices  = (const int*)d_in[1];
    const float* weight   = (const float*)d_in[2];
    float*       out      = (float*)d_out;

    const int N = in_sizes[1] / 3;  // indices is (N,3)

    // 1) zero the dense output grid (~83.6 MB)
    sc_zero_kernel<<<2048, 256, 0, stream>>>(out, (long)out_size);

    // 2) WMMA compute + atomic scatter; 128 points per 256-thread block
    const int blocks = (N + 127) / 128;
    sc_conv_wmma_kernel<<<blocks, 256, 0, stream>>>(features, indices, weight, out, N);
}